// NoisyNetwork_24103356465499
// MI455X (gfx1250) — compile-verified
//
#include <hip/hip_runtime.h>
#include <hip/hip_bf16.h>

// Problem constants (match reference)
#define B_DIM   16384
#define IN_DIM  400
#define HID     24
#define OUT_DIM 4096
#define KMOV    64

typedef __attribute__((ext_vector_type(2))) float v2f;
typedef __attribute__((ext_vector_type(4))) float v4f;
typedef __attribute__((ext_vector_type(8))) float v8f;

#define LDS_STRIDE      26      // 24 + pad, keeps float2 loads 8B aligned & bank-spread
#define WAVES_PER_BLOCK 2
#define ROWS_PER_WAVE   16
#define THREADS         (WAVES_PER_BLOCK * 32)
#define ROWS_PER_BLOCK  (WAVES_PER_BLOCK * ROWS_PER_WAVE)

__launch_bounds__(THREADS)
__global__ void noisy_fused_kernel(
    const float* __restrict__ x,
    const int*   __restrict__ pm,
    const float* __restrict__ W1,
    const float* __restrict__ b1,
    const float* __restrict__ mu_w2,    const float* __restrict__ mu_b2,
    const float* __restrict__ sigma_w2, const float* __restrict__ sigma_b2,
    const float* __restrict__ eps_w2,   const float* __restrict__ eps_b2,
    const float* __restrict__ mu_w3,    const float* __restrict__ mu_b3,
    const float* __restrict__ sigma_w3, const float* __restrict__ sigma_b3,
    const float* __restrict__ eps_w3,   const float* __restrict__ eps_b3,
    float* __restrict__ out)
{
    __shared__ __align__(16) float h1s[WAVES_PER_BLOCK][16 * LDS_STRIDE];
    __shared__ __align__(16) float h2s[WAVES_PER_BLOCK][16 * LDS_STRIDE];

    const int lane = threadIdx.x & 31;
    const int wave = threadIdx.x >> 5;
    const int half = lane >> 4;      // K sub-group within the wave (ISA A/B layout)
    const int l16  = lane & 15;      // M (for A) / N (for B,C,D)
    const int rowBase = (blockIdx.x * WAVES_PER_BLOCK + wave) * ROWS_PER_WAVE;

    const int  n1     = 16 + l16;            // second N tile column (valid if < 24)
    const bool n1ok   = (n1 < HID);
    const int  n1c    = n1ok ? n1 : 0;       // clamped (safe address) column
    const v2f  vzero  = {0.0f, 0.0f};
    const float NEG_INF = -__builtin_huge_valf();

    // ---------------- GEMM1: h1 = relu(x @ W1^T + b1) ----------------
    // A: 16x4 f32 tile per step; lane layout: VGPR0 = K(2*half), VGPR1 = K(2*half+1)
    v8f acc0 = {}; v8f acc1 = {};
    {
        const float* xrow  = x  + (size_t)(rowBase + l16) * IN_DIM + 2 * half;
        const float* w1r0  = W1 + (size_t)l16 * IN_DIM + 2 * half;       // N tile 0
        const float* w1r1  = W1 + (size_t)n1c * IN_DIM + 2 * half;       // N tile 1 (clamped)
        for (int k0 = 0; k0 < IN_DIM; k0 += 4) {
            v2f a  = *(const v2f*)(xrow + k0);
            v2f b0 = *(const v2f*)(w1r0 + k0);
            v2f bt = *(const v2f*)(w1r1 + k0);
            v2f b1v = n1ok ? bt : vzero;
            acc0 = __builtin_amdgcn_wmma_f32_16x16x4_f32(false, a, false, b0,
                                                         (short)0, acc0, false, false);
            acc1 = __builtin_amdgcn_wmma_f32_16x16x4_f32(false, a, false, b1v,
                                                         (short)0, acc1, false, false);
        }
    }

    // bias + relu, store h1 tile to this wave's LDS region (row-major [16][24])
    float* h1w = h1s[wave];
    {
        const float bn0 = b1[l16];
        const float bn1 = b1[n1c];
        #pragma unroll
        for (int i = 0; i < 8; ++i) {
            const int m = i + 8 * half;           // C layout: vgpr i, lane group -> row
            float v0 = acc0[i] + bn0; v0 = v0 > 0.0f ? v0 : 0.0f;
            h1w[m * LDS_STRIDE + l16] = v0;
            if (n1ok) {
                float v1 = acc1[i] + bn1; v1 = v1 > 0.0f ? v1 : 0.0f;
                h1w[m * LDS_STRIDE + n1] = v1;
            }
        }
    }
    __syncthreads();

    // ---------------- GEMM2: h2 = relu(h1 @ w2^T + b2'), w2 = mu + sigma*eps ----------------
    v8f d0 = {}; v8f d1 = {};
    {
        #pragma unroll
        for (int kk = 0; kk < HID; kk += 4) {
            const int k = kk + 2 * half;
            v2f a = *(const v2f*)(h1w + l16 * LDS_STRIDE + k);
            // B tile 0: w2t[k][n] = w2[n][k], n = l16 (< 16 < 24 always valid)
            v2f mu0 = *(const v2f*)(mu_w2    + l16 * HID + k);
            v2f sg0 = *(const v2f*)(sigma_w2 + l16 * HID + k);
            v2f ep0 = *(const v2f*)(eps_w2   + l16 * HID + k);
            v2f b0  = mu0 + sg0 * ep0;
            // B tile 1: n = 16 + l16 (valid only when < 24; clamped address, masked value)
            v2f mu1 = *(const v2f*)(mu_w2    + n1c * HID + k);
            v2f sg1 = *(const v2f*)(sigma_w2 + n1c * HID + k);
            v2f ep1 = *(const v2f*)(eps_w2   + n1c * HID + k);
            v2f bt  = mu1 + sg1 * ep1;
            v2f b1v = n1ok ? bt : vzero;
            d0 = __builtin_amdgcn_wmma_f32_16x16x4_f32(false, a, false, b0,
                                                       (short)0, d0, false, false);
            d1 = __builtin_amdgcn_wmma_f32_16x16x4_f32(false, a, false, b1v,
                                                       (short)0, d1, false, false);
        }
    }

    // bias + relu, store h2 tile to LDS
    float* h2w = h2s[wave];
    {
        const float c0 = mu_b2[l16] + sigma_b2[l16] * eps_b2[l16];
        const float c1 = mu_b2[n1c] + sigma_b2[n1c] * eps_b2[n1c];
        #pragma unroll
        for (int i = 0; i < 8; ++i) {
            const int m = i + 8 * half;
            float v0 = d0[i] + c0; v0 = v0 > 0.0f ? v0 : 0.0f;
            h2w[m * LDS_STRIDE + l16] = v0;
            if (n1ok) {
                float v1 = d1[i] + c1; v1 = v1 > 0.0f ? v1 : 0.0f;
                h2w[m * LDS_STRIDE + n1] = v1;
            }
        }
    }
    __syncthreads();

    // ---------------- Fill this wave's 16 output rows with -inf (NT b128 stores) ----------------
    {
        v4f ninf4 = {NEG_INF, NEG_INF, NEG_INF, NEG_INF};
        v4f* obase = (v4f*)(out + (size_t)rowBase * OUT_DIM);
        const int n4 = ROWS_PER_WAVE * OUT_DIM / 4;   // 16384 float4 per wave
        #pragma unroll 4
        for (int i = lane; i < n4; i += 32)
            __builtin_nontemporal_store(ninf4, obase + i);
    }

    // Same-wave store->store ordering fence before overwriting legal positions.
    asm volatile("s_wait_storecnt 0x0" ::: "memory");

    // ---------------- Gathered layer 3: only legal-move logits ----------------
    #pragma unroll 1
    for (int r = 0; r < ROWS_PER_WAVE; ++r) {
        float h[HID];
        #pragma unroll
        for (int c = 0; c < HID; ++c) h[c] = h2w[r * LDS_STRIDE + c];   // LDS broadcast

        const int row = rowBase + r;
        const int* pmr = pm + (size_t)row * KMOV;
        float* orow = out + (size_t)row * OUT_DIM;

        #pragma unroll
        for (int jj = 0; jj < KMOV; jj += 32) {
            const int idx = pmr[jj + lane];
            // w3 row = mu + sigma*eps; 24 floats = 96B -> 16B aligned float4 loads
            const v4f* mw = (const v4f*)(mu_w3    + (size_t)idx * HID);
            const v4f* sw = (const v4f*)(sigma_w3 + (size_t)idx * HID);
            const v4f* ew = (const v4f*)(eps_w3   + (size_t)idx * HID);
            float acc = mu_b3[idx] + sigma_b3[idx] * eps_b3[idx];
            #pragma unroll
            for (int q = 0; q < HID / 4; ++q) {
                v4f w = mw[q] + sw[q] * ew[q];
                acc += h[4*q + 0] * w[0];
                acc += h[4*q + 1] * w[1];
                acc += h[4*q + 2] * w[2];
                acc += h[4*q + 3] * w[3];
            }
            // reference: filtered==0 -> -inf (covers logit==+-0 at legal positions too)
            orow[idx] = (acc == 0.0f) ? NEG_INF : acc;
        }
    }
}

extern "C" void kernel_launch(void* const* d_in, const int* in_sizes, int n_in,
                              void* d_out, int out_size, void* d_ws, size_t ws_size,
                              hipStream_t stream) {
    (void)in_sizes; (void)n_in; (void)out_size; (void)d_ws; (void)ws_size;
    const float* x        = (const float*)d_in[0];
    const int*   pm       = (const int*)  d_in[1];
    const float* W1       = (const float*)d_in[2];
    const float* b1       = (const float*)d_in[3];
    const float* mu_w2    = (const float*)d_in[4];
    const float* mu_b2    = (const float*)d_in[5];
    const float* sigma_w2 = (const float*)d_in[6];
    const float* sigma_b2 = (const float*)d_in[7];
    const float* eps_w2   = (const float*)d_in[8];
    const float* eps_b2   = (const float*)d_in[9];
    const float* mu_w3    = (const float*)d_in[10];
    const float* mu_b3    = (const float*)d_in[11];
    const float* sigma_w3 = (const float*)d_in[12];
    const float* sigma_b3 = (const float*)d_in[13];
    const float* eps_w3   = (const float*)d_in[14];
    const float* eps_b3   = (const float*)d_in[15];

    dim3 grid(B_DIM / ROWS_PER_BLOCK);   // 512 blocks
    dim3 block(THREADS);                 // 64 threads = 2 waves (wave32)
    noisy_fused_kernel<<<grid, block, 0, stream>>>(
        x, pm, W1, b1,
        mu_w2, mu_b2, sigma_w2, sigma_b2, eps_w2, eps_b2,
        mu_w3, mu_b3, sigma_w3, sigma_b3, eps_w3, eps_b3,
        (float*)d_out);
}